// VectorQuantizer_1494648619464
// MI455X (gfx1250) — compile-verified
//
#include <hip/hip_runtime.h>
#include <hip/hip_bf16.h>

// Vector Quantizer for MI455X (gfx1250).
// dist GEMM runs on the fp32 matrix pipe: V_WMMA_F32_16X16X4_F32
// (full fp32 A/B precision -> argmin matches the fp32 JAX reference).

typedef __attribute__((ext_vector_type(2))) float v2f;
typedef __attribute__((ext_vector_type(8))) float v8f;

#if defined(__has_builtin)
#  if __has_builtin(__builtin_amdgcn_sched_barrier)
#    define SCHED_FENCE() __builtin_amdgcn_sched_barrier(0)
#  endif
#endif
#ifndef SCHED_FENCE
#  define SCHED_FENCE() asm volatile("" ::: "memory")
#endif

#define EMB_DIM        64
#define NUM_CODES      512
#define CHUNK          128        // codes staged in LDS at a time
#define LDS_STRIDE     68         // padded row stride (floats) -> no LDS bank conflicts
#define N_ROWS         65536      // 64*32*32
#define ROWS_PER_WAVE  16
#define WAVES_PER_BLK  8
#define NUM_BLOCKS     (N_ROWS / (ROWS_PER_WAVE * WAVES_PER_BLK))   // 512

// d_ws float layout: [0..511] codebook squared norms, [512..1023] per-block loss partials.

__global__ __launch_bounds__(512) void vq_prep(const float* __restrict__ cb,
                                               float* __restrict__ ws) {
  int k = blockIdx.x * blockDim.x + threadIdx.x;
  if (k < NUM_CODES) {
    const float* row = cb + (size_t)k * EMB_DIM;
    float s = 0.0f;
#pragma unroll
    for (int d = 0; d < EMB_DIM; ++d) s = fmaf(row[d], row[d], s);
    ws[k] = s;
  }
}

__global__ __launch_bounds__(256) void vq_main(const float* __restrict__ x,
                                               const float* __restrict__ cb,
                                               float* __restrict__ ws,
                                               float* __restrict__ out) {
  __shared__ __align__(16) float cbLds[CHUNK * LDS_STRIDE];   // ~34 KB
  __shared__ float normLds[CHUNK];
  __shared__ float waveLoss[WAVES_PER_BLK];

  const int tid   = threadIdx.x;
  const int lane  = tid & 31;
  const int wave  = tid >> 5;
  const int laneM = lane & 15;
  const int hi    = lane >> 4;        // 0: K-off 0 / rows 0..7 ; 1: K-off 2 / rows 8..15
  const int colOff = hi * 2;

  const int tile    = blockIdx.x * WAVES_PER_BLK + wave;
  const int rowBase = tile * ROWS_PER_WAVE;

  // A fragments: 16x4 fp32 per step. Lane layout (ISA 7.12.2): lanes 0-15 hold
  // K=0,1 of row M=lane; lanes 16-31 hold K=2,3 of row M=lane-16.
  const float* xrow = x + (size_t)(rowBase + laneM) * EMB_DIM + colOff;
  v2f a[16];
#pragma unroll
  for (int kk = 0; kk < 16; ++kk) a[kk] = *(const v2f*)(xrow + kk * 4);

  float minD[8];
  int   minI[8];
#pragma unroll
  for (int i = 0; i < 8; ++i) { minD[i] = 3.4e38f; minI[i] = 0; }

  for (int chunk = 0; chunk < NUM_CODES / CHUNK; ++chunk) {
    __syncthreads();   // previous chunk fully consumed
    {
      // 256 threads stage 128 codes x 64 floats: two threads per code row.
      const int code = tid >> 1;
      const int h2   = (tid & 1) * 32;
      const float4* src = (const float4*)(cb + (size_t)(chunk * CHUNK + code) * EMB_DIM + h2);
      float4* dst = (float4*)(cbLds + code * LDS_STRIDE + h2);
#pragma unroll
      for (int j = 0; j < 8; ++j) dst[j] = src[j];
      if (tid < CHUNK) normLds[tid] = ws[chunk * CHUNK + tid];
    }
    __syncthreads();

#pragma unroll 1
    for (int ct = 0; ct < CHUNK / 16; ++ct) {
      const int codeLocal = ct * 16 + laneM;
      const float* brow = cbLds + codeLocal * LDS_STRIDE + colOff;

      // Issue ALL B-fragment LDS loads (8x ds_load_2addr_b64, in-order
      // returns) before any WMMA; the sched fence stops the backend from
      // sinking loads back next to their consumers. Waitcnt insertion can
      // then use partial s_wait_dscnt so the WMMA chain overlaps the tail
      // of the LDS latency.
      v2f b[16];
#pragma unroll
      for (int kk = 0; kk < 16; ++kk) b[kk] = *(const v2f*)(brow + kk * 4);
      const float nrm = normLds[codeLocal];

      SCHED_FENCE();

      v8f c = {};
#pragma unroll
      for (int kk = 0; kk < 16; ++kk)
        c = __builtin_amdgcn_wmma_f32_16x16x4_f32(false, a[kk], false, b[kk],
                                                  (short)0, c, false, false);

      const int codeGlobal = chunk * CHUNK + codeLocal;
#pragma unroll
      for (int i = 0; i < 8; ++i) {
        float dist = fmaf(-2.0f, c[i], nrm);    // ||e||^2 - 2 z.e  (||z||^2 is argmin-invariant)
        if (dist < minD[i]) { minD[i] = dist; minI[i] = codeGlobal; }
      }
    }
  }

  // Butterfly argmin across the 16 lanes holding each row's columns
  // (masks 1..8 stay inside each 16-lane half). Tie-break: smaller index (JAX argmin).
#pragma unroll
  for (int off = 1; off < 16; off <<= 1) {
#pragma unroll
    for (int i = 0; i < 8; ++i) {
      float od = __shfl_xor(minD[i], off, 32);
      int   oi = __shfl_xor(minI[i], off, 32);
      if (od < minD[i] || (od == minD[i] && oi < minI[i])) { minD[i] = od; minI[i] = oi; }
    }
  }

  // Output layout: [quantized 4194304][loss 1][idx 65536]
  float* qOut   = out;
  float* idxOut = out + (size_t)N_ROWS * EMB_DIM + 1;

  float lossAcc = 0.0f;
#pragma unroll 1
  for (int i = 0; i < 8; ++i) {
    const int row  = rowBase + hi * 8 + i;   // lanes 0-15: rows 0..7, lanes 16-31: rows 8..15
    const int best = minI[i];                // uniform within each 16-lane half
    if (laneM == 0) idxOut[row] = (float)best;
    const float4 q  = ((const float4*)(cb + (size_t)best * EMB_DIM))[laneM];
    const float4 xv = ((const float4*)(x  + (size_t)row  * EMB_DIM))[laneM];
    ((float4*)(qOut + (size_t)row * EMB_DIM))[laneM] = q;
    float d0 = q.x - xv.x, d1 = q.y - xv.y, d2 = q.z - xv.z, d3 = q.w - xv.w;
    lossAcc += d0 * d0 + d1 * d1 + d2 * d2 + d3 * d3;
  }

#pragma unroll
  for (int off = 16; off >= 1; off >>= 1) lossAcc += __shfl_xor(lossAcc, off, 32);
  if (lane == 0) waveLoss[wave] = lossAcc;
  __syncthreads();
  if (tid == 0) {
    float s = 0.0f;
#pragma unroll
    for (int w = 0; w < WAVES_PER_BLK; ++w) s += waveLoss[w];
    ws[NUM_CODES + blockIdx.x] = s;          // deterministic: no atomics
  }
}

__global__ void vq_finalize(const float* __restrict__ ws, float* __restrict__ out) {
  if (threadIdx.x == 0 && blockIdx.x == 0) {
    float s = 0.0f;
    for (int b = 0; b < NUM_BLOCKS; ++b) s += ws[NUM_CODES + b];
    // loss = q_latent + 0.25*e_latent = 1.25 * mean((q-x)^2), mean over 65536*64 elems
    out[(size_t)N_ROWS * EMB_DIM] = 1.25f * s * (1.0f / 4194304.0f);
  }
}

extern "C" void kernel_launch(void* const* d_in, const int* in_sizes, int n_in,
                              void* d_out, int out_size, void* d_ws, size_t ws_size,
                              hipStream_t stream) {
  const float* x  = (const float*)d_in[0];   // [64,32,32,64] f32
  const float* cb = (const float*)d_in[1];   // [512,64] f32
  float* out = (float*)d_out;
  float* ws  = (float*)d_ws;                 // needs (512+512)*4 = 4 KB

  vq_prep<<<1, 512, 0, stream>>>(cb, ws);
  vq_main<<<NUM_BLOCKS, 256, 0, stream>>>(x, cb, ws, out);
  vq_finalize<<<1, 64, 0, stream>>>(ws, out);
}